// SACRSN_v55_23536420782583
// MI455X (gfx1250) — compile-verified
//
#include <hip/hip_runtime.h>
#include <math.h>

// Problem constants (match reference)
#define B_     512
#define D_     512
#define DD_    1024     // 2*D
#define SLOTS_ 32
#define KC_    128      // VQ codes
#define RITER_ 8
#define VV_    50257
#define ANCH_  32
#define NCLIN_ 5120     // 5 complex-linears * 1024 outputs

typedef __attribute__((ext_vector_type(2))) float v2f;
typedef __attribute__((ext_vector_type(8))) float v8f;

__device__ __forceinline__ float sigf(float x) { return 1.f / (1.f + expf(-x)); }

__device__ __forceinline__ float waveReduceSum(float v) {
#pragma unroll
  for (int o = 16; o; o >>= 1) v += __shfl_xor(v, o, 32);
  return v;
}

__device__ __forceinline__ float blockReduceSum(float v, float* red) {
  const int lane = threadIdx.x & 31, wave = threadIdx.x >> 5;
  v = waveReduceSum(v);
  if (lane == 0) red[wave] = v;
  __syncthreads();
  const int nw = blockDim.x >> 5;
  float s = (threadIdx.x < nw) ? red[threadIdx.x] : 0.f;
  if (wave == 0) { s = waveReduceSum(s); if (lane == 0) red[0] = s; }
  __syncthreads();
  float r = red[0];
  __syncthreads();
  return r;
}

// Softmax over 32 shared entries, executed by one full wave.
__device__ __forceinline__ void waveSoftmax32(float* arr) {
  const int lane = threadIdx.x & 31;
  float v = arr[lane];
  float m = v;
#pragma unroll
  for (int o = 16; o; o >>= 1) m = fmaxf(m, __shfl_xor(m, o, 32));
  float e = expf(v - m);
  float s = e;
#pragma unroll
  for (int o = 16; o; o >>= 1) s += __shfl_xor(s, o, 32);
  arr[lane] = e / s;
}

// ---------------------------------------------------------------------------
// Generic fp32 WMMA GEMM: C[M,N] = A[M,K] @ W[N,K]^T + bias[N]
// Block: 256 threads (8 waves). Block tile 64(M) x 128(N).
// Each wave: 16-col slice x 4 M-subtiles = 4 independent v8f accumulator
// chains (B fragment reused 4x, XDL latency hidden).
// Both tiles streamed straight into LDS with global_load_async_to_lds_b128
// (ASYNCcnt) using DOUBLE BUFFERING: chunk i+1 loads overlap chunk i WMMAs.
// ---------------------------------------------------------------------------
#define KCH 64
#define LDP (KCH + 4)   // pad so fragment reads land in distinct banks

__device__ __forceinline__ void issue_tile_async(
    const float* __restrict__ A, const float* __restrict__ W,
    int N, int K, int m0, int n0, int k0, int tid,
    float (*lAbuf)[LDP], float (*lWbuf)[LDP])
{
  const int rBase = tid >> 4;
  const int c4 = (tid & 15) << 2;
#pragma unroll
  for (int pass = 0; pass < 4; ++pass) {            // A tile: 64 x 64
    const int r = rBase + pass * 16;
    const float* src = A + (size_t)(m0 + r) * K + k0 + c4;
    const unsigned int ldsOff = (unsigned int)(unsigned long long)&lAbuf[r][c4];
    asm volatile("global_load_async_to_lds_b128 %0, %1, off"
                 :: "v"(ldsOff), "v"((unsigned long long)(const void*)src)
                 : "memory");
  }
#pragma unroll
  for (int itr = 0; itr < 8; ++itr) {               // W tile: 128 x 64
    const int r = rBase + itr * 16;
    int n = n0 + r; if (n >= N) n = N - 1;          // clamp (EXEC stays uniform)
    const float* src = W + (size_t)n * K + k0 + c4;
    const unsigned int ldsOff = (unsigned int)(unsigned long long)&lWbuf[r][c4];
    asm volatile("global_load_async_to_lds_b128 %0, %1, off"
                 :: "v"(ldsOff), "v"((unsigned long long)(const void*)src)
                 : "memory");
    if (k0 + 2 * KCH < K) __builtin_prefetch(src + 2 * KCH, 0, 1); // warm L2
  }
}

__global__ __launch_bounds__(256, 2) void gemm_f32_wmma(
    const float* __restrict__ A, const float* __restrict__ W,
    const float* __restrict__ bias, float* __restrict__ C,
    int M, int N, int K)
{
  __shared__ float lA[2][64][LDP];
  __shared__ float lW[2][128][LDP];
  const int tid  = threadIdx.x;
  const int wave = tid >> 5, lane = tid & 31;
  const int m0 = blockIdx.x * 64;
  const int n0 = blockIdx.y * 128;

  v8f acc0 = {0.f,0.f,0.f,0.f,0.f,0.f,0.f,0.f};
  v8f acc1 = acc0, acc2 = acc0, acc3 = acc0;

  const int ml = lane & 15;                 // M (A frag) / N (B frag) index
  const int hi = (lane >= 16) ? 2 : 0;      // K sub-offset per ISA 16x4 layout
  const int nl = wave * 16 + ml;            // local N column for this wave

  // Prologue: stage chunk 0 into buffer 0.
  issue_tile_async(A, W, N, K, m0, n0, 0, tid, lA[0], lW[0]);
  asm volatile("s_wait_asynccnt 0x0" ::: "memory");
  __syncthreads();

  const int nCh = K / KCH;
  for (int ch = 0; ch < nCh; ++ch) {
    const int cur = ch & 1;
    // Overlap: issue async loads for the next chunk into the other buffer.
    if (ch + 1 < nCh)
      issue_tile_async(A, W, N, K, m0, n0, (ch + 1) * KCH, tid,
                       lA[cur ^ 1], lW[cur ^ 1]);

    float (*cA)[LDP] = lA[cur];
    float (*cW)[LDP] = lW[cur];
#pragma unroll
    for (int k = 0; k < KCH; k += 4) {
      v2f bf;
      bf[0] = cW[nl][k + hi];  bf[1] = cW[nl][k + hi + 1];
      v2f a0, a1, a2, a3;
      a0[0] = cA[ 0 + ml][k + hi]; a0[1] = cA[ 0 + ml][k + hi + 1];
      a1[0] = cA[16 + ml][k + hi]; a1[1] = cA[16 + ml][k + hi + 1];
      a2[0] = cA[32 + ml][k + hi]; a2[1] = cA[32 + ml][k + hi + 1];
      a3[0] = cA[48 + ml][k + hi]; a3[1] = cA[48 + ml][k + hi + 1];
      acc0 = __builtin_amdgcn_wmma_f32_16x16x4_f32(false, a0, false, bf, (short)0, acc0, false, false);
      acc1 = __builtin_amdgcn_wmma_f32_16x16x4_f32(false, a1, false, bf, (short)0, acc1, false, false);
      acc2 = __builtin_amdgcn_wmma_f32_16x16x4_f32(false, a2, false, bf, (short)0, acc2, false, false);
      acc3 = __builtin_amdgcn_wmma_f32_16x16x4_f32(false, a3, false, bf, (short)0, acc3, false, false);
    }
    // Next-chunk loads had the whole compute phase to land; drain + barrier.
    asm volatile("s_wait_asynccnt 0x0" ::: "memory");
    __syncthreads();
  }

  // C/D layout: VGPR r -> lanes 0-15: (M=r, N=lane); lanes 16-31: (M=r+8, N=lane-16)
  const int n = n0 + nl;
  if (n < N) {
    const float bv = bias ? bias[n] : 0.f;
    const int mb = m0 + ((lane >= 16) ? 8 : 0);
#pragma unroll
    for (int r = 0; r < 8; ++r) {
      C[(size_t)(mb +  0 + r) * N + n] = acc0[r] + bv;
      C[(size_t)(mb + 16 + r) * N + n] = acc1[r] + bv;
      C[(size_t)(mb + 32 + r) * N + n] = acc2[r] + bv;
      C[(size_t)(mb + 48 + r) * N + n] = acc3[r] + bv;
    }
  }
}

// ---------------------------------------------------------------------------
// Prep: gbuf[b] = [alpha*gw_r+(1-a)*x_r | alpha*gw_i+(1-a)*x_i]
// ---------------------------------------------------------------------------
__global__ __launch_bounds__(256) void prep_input(
    const float* __restrict__ xR, const float* __restrict__ xI,
    const float* __restrict__ gwR, const float* __restrict__ gwI,
    const float* __restrict__ inputGate, float* __restrict__ gbuf)
{
  const int idx = blockIdx.x * 256 + threadIdx.x;
  if (idx >= B_ * D_) return;
  const float alpha = sigf(inputGate[0]);
  const int b = idx / D_, d = idx % D_;
  gbuf[(size_t)b * DD_ + d]       = alpha * gwR[idx] + (1.f - alpha) * xR[idx];
  gbuf[(size_t)b * DD_ + D_ + d]  = alpha * gwI[idx] + (1.f - alpha) * xI[idx];
}

// ---------------------------------------------------------------------------
// Prep: combined complex-linear weight (row n of Wc is output col, length 2D)
//   out_r = cr@Wr.T - ci@Wi.T + (br-bi) ; out_i = cr@Wi.T + ci@Wr.T + (br+bi)
// ---------------------------------------------------------------------------
__global__ __launch_bounds__(256) void prep_wclin(
    const float* __restrict__ Wr, const float* __restrict__ br,
    const float* __restrict__ Wi, const float* __restrict__ bi,
    float* __restrict__ Wc, float* __restrict__ bc)
{
  const int j = blockIdx.x;       // 0..1023 output column within this clin
  const int t = threadIdx.x;
  float* wrow = Wc + (size_t)j * DD_;
  if (j < D_) {
    for (int k = t; k < D_; k += 256) wrow[k]       =  Wr[(size_t)j * D_ + k];
    for (int k = t; k < D_; k += 256) wrow[D_ + k]  = -Wi[(size_t)j * D_ + k];
    if (t == 0) bc[j] = br[j] - bi[j];
  } else {
    const int j2 = j - D_;
    for (int k = t; k < D_; k += 256) wrow[k]       =  Wi[(size_t)j2 * D_ + k];
    for (int k = t; k < D_; k += 256) wrow[D_ + k]  =  Wr[(size_t)j2 * D_ + k];
    if (t == 0) bc[j] = br[j2] + bi[j2];
  }
}

__global__ __launch_bounds__(128) void prep_enorm(
    const float* __restrict__ E, float* __restrict__ enorm)
{
  const int k = threadIdx.x;
  const float* e = E + (size_t)k * DD_;
  float s = 0.f;
  for (int i = 0; i < DD_; ++i) s += e[i] * e[i];
  enorm[k] = s;
}

// ---------------------------------------------------------------------------
// LayerNorm both halves of gbuf row -> zf row  ([cr | ci], contiguous 1024)
// ---------------------------------------------------------------------------
__global__ __launch_bounds__(256) void ln2_kernel(
    const float* __restrict__ gbuf,
    const float* __restrict__ g_r, const float* __restrict__ b_r,
    const float* __restrict__ g_i, const float* __restrict__ b_i,
    float* __restrict__ zf)
{
  __shared__ float red[8];
  const int b = blockIdx.x, t = threadIdx.x;
  const float* row = gbuf + (size_t)b * DD_;
#pragma unroll
  for (int half = 0; half < 2; ++half) {
    const float* x  = row + half * D_;
    const float* gg = half ? g_i : g_r;
    const float* bb = half ? b_i : b_r;
    const float v0 = x[t], v1 = x[t + 256];
    const float s  = blockReduceSum(v0 + v1, red);
    const float mu = s * (1.f / D_);
    const float d0 = v0 - mu, d1 = v1 - mu;
    const float s2 = blockReduceSum(d0 * d0 + d1 * d1, red);
    const float inv = rsqrtf(s2 * (1.f / D_) + 1e-5f);
    float* zr = zf + (size_t)b * DD_ + half * D_;
    zr[t]       = d0 * inv * gg[t]       + bb[t];
    zr[t + 256] = d1 * inv * gg[t + 256] + bb[t + 256];
  }
}

// ---------------------------------------------------------------------------
// VQ: argmin_k (||E_k||^2 - 2 zf.E_k) ; straight-through forward value = E[idx]
// ---------------------------------------------------------------------------
__global__ __launch_bounds__(128) void vq_kernel(
    const float* __restrict__ zf, const float* __restrict__ E,
    const float* __restrict__ enorm, float* __restrict__ vq)
{
  __shared__ float zrow[DD_];
  __shared__ float dist[KC_];
  __shared__ int   didx[KC_];
  const int b = blockIdx.x, t = threadIdx.x;
  for (int i = t; i < DD_; i += 128) zrow[i] = zf[(size_t)b * DD_ + i];
  __syncthreads();
  const float* Ek = E + (size_t)t * DD_;
  float dot = 0.f;
  for (int i = 0; i < DD_; ++i) dot += zrow[i] * Ek[i];
  dist[t] = enorm[t] - 2.f * dot;
  didx[t] = t;
  __syncthreads();
  for (int s = 64; s > 0; s >>= 1) {
    if (t < s) {
      if (dist[t + s] < dist[t] ||
          (dist[t + s] == dist[t] && didx[t + s] < didx[t])) {
        dist[t] = dist[t + s]; didx[t] = didx[t + s];
      }
    }
    __syncthreads();
  }
  const float* Ei = E + (size_t)didx[0] * DD_;
  for (int i = t; i < DD_; i += 128) vq[(size_t)b * DD_ + i] = Ei[i];
}

// ---------------------------------------------------------------------------
// Per-row fused kernel: qk gate, slot attention, write gate/address, palettes,
// arbitration, final mix -> new gbuf row + eff[b,s]
// ---------------------------------------------------------------------------
__global__ __launch_bounds__(256) void row_kernel(
    const float* __restrict__ zf, const float* __restrict__ out5,
    const float* __restrict__ vq,
    const float* __restrict__ memR, const float* __restrict__ memI,
    const float* __restrict__ gateW, const float* __restrict__ gateB,
    const float* __restrict__ addrW, const float* __restrict__ addrB,
    const float* __restrict__ visP, const float* __restrict__ audP,
    const float* __restrict__ arbW, const float* __restrict__ arbB,
    float* __restrict__ gbuf, float* __restrict__ eff)
{
  __shared__ float red[8];
  __shared__ float zrow[DD_];
  __shared__ float mR[D_], mI[D_];
  __shared__ float simv[SLOTS_], attn[SLOTS_], wwv[SLOTS_], va[ANCH_], aa[ANCH_];
  __shared__ float arb[5];
  __shared__ float scal[2];

  const int b = blockIdx.x, t = threadIdx.x;
  const int lane = t & 31, wave = t >> 5;
  const float* o5 = out5 + (size_t)b * NCLIN_;   // [q|k|v|proj|critic] each 1024
  const float* zg = zf + (size_t)b * DD_;

  for (int i = t; i < DD_; i += 256) zrow[i] = zg[i];
  __syncthreads();

  // gate = sigmoid(sum(q .* k)) over full complex vector (layout makes it one dot)
  float part = 0.f;
  for (int i = t; i < DD_; i += 256) part += o5[i] * o5[DD_ + i];
  const float qk = blockReduceSum(part, red);
  // wg = sigmoid(zf . gate_W + gate_b)
  part = 0.f;
  for (int i = t; i < DD_; i += 256) part += zrow[i] * gateW[i];
  const float wgs = blockReduceSum(part, red);
  if (t == 0) { scal[0] = sigf(qk); scal[1] = sigf(wgs + gateB[0]); }
  __syncthreads();

  const size_t mb = (size_t)b * SLOTS_ * D_;
  // Wave-per-row dot products: sim, ww, vis, aud
  for (int s = wave; s < SLOTS_; s += 8) {
    const float* mrw = memR + mb + (size_t)s * D_;
    const float* miw = memI + mb + (size_t)s * D_;
    float p = 0.f;
    for (int i = lane; i < D_; i += 32) p += mrw[i] * zrow[i] + miw[i] * zrow[D_ + i];
    p = waveReduceSum(p);
    if (lane == 0) simv[s] = p;

    const float* aw = addrW + (size_t)s * DD_;
    float q = 0.f;
    for (int i = lane; i < DD_; i += 32) q += aw[i] * zrow[i];
    q = waveReduceSum(q);
    if (lane == 0) wwv[s] = q + addrB[s];

    const float* vp = visP + (size_t)s * DD_;
    const float* ap = audP + (size_t)s * DD_;
    const float* zs = o5 + 3 * DD_;   // zsf = [proj_r | proj_i]
    float pv = 0.f, pa = 0.f;
    for (int i = lane; i < DD_; i += 32) { const float z = zs[i]; pv += vp[i] * z; pa += ap[i] * z; }
    pv = waveReduceSum(pv); pa = waveReduceSum(pa);
    if (lane == 0) { va[s] = pv; aa[s] = pa; }
  }
  // arbitration logits (5 rows), one wave each
  if (wave < 5) {
    const float* awr = arbW + (size_t)wave * DD_;
    float p = 0.f;
    for (int i = lane; i < DD_; i += 32) p += awr[i] * zrow[i];
    p = waveReduceSum(p);
    if (lane == 0) arb[wave] = p + arbB[wave];
  }
  __syncthreads();

  if (wave == 0) { // softmax(sim) -> attn
    float v = simv[lane], m = v;
#pragma unroll
    for (int o = 16; o; o >>= 1) m = fmaxf(m, __shfl_xor(m, o, 32));
    float e = expf(v - m), s = e;
#pragma unroll
    for (int o = 16; o; o >>= 1) s += __shfl_xor(s, o, 32);
    attn[lane] = e / s;
  }
  if (wave == 1) waveSoftmax32(wwv);
  if (wave == 2) waveSoftmax32(va);
  if (wave == 3) waveSoftmax32(aa);
  if (t == 128) { // softmax over 5 arbitration gates (serial, tiny)
    float m = arb[0];
    for (int j = 1; j < 5; ++j) m = fmaxf(m, arb[j]);
    float s = 0.f;
    for (int j = 0; j < 5; ++j) { arb[j] = expf(arb[j] - m); s += arb[j]; }
    for (int j = 0; j < 5; ++j) arb[j] /= s;
  }
  __syncthreads();

  if (wave == 0 && lane < SLOTS_) eff[b * SLOTS_ + lane] = scal[1] * wwv[lane];

  // m_r/m_i = attn @ mem (old memory)
  for (int d = t; d < D_; d += 256) {
    float ar = 0.f, ai = 0.f;
    for (int s = 0; s < SLOTS_; ++s) {
      const float a = attn[s];
      ar += a * memR[mb + (size_t)s * D_ + d];
      ai += a * memI[mb + (size_t)s * D_ + d];
    }
    mR[d] = ar; mI[d] = ai;
  }

  // Final mix -> new gr/gi in gbuf
  const float gate = scal[0];
  const float* vqr = vq + (size_t)b * DD_;
  for (int d = t; d < D_; d += 256) {
    const float crd = zrow[d], cid = zrow[D_ + d];
    const float g_r = o5[2 * DD_ + d] * gate;
    const float g_i = o5[2 * DD_ + D_ + d] * gate;
    const float v_r = vqr[d], v_i = vqr[D_ + d];
    float visr = 0.f, visi = 0.f, audr = 0.f, audi = 0.f;
    for (int a = 0; a < ANCH_; ++a) {
      const float wa = va[a], ba = aa[a];
      visr += wa * visP[(size_t)a * DD_ + d];
      visi += wa * visP[(size_t)a * DD_ + D_ + d];
      audr += ba * audP[(size_t)a * DD_ + d];
      audi += ba * audP[(size_t)a * DD_ + D_ + d];
    }
    const float s_r = visr - audi, s_i = visi + audr;
    const float czr = o5[4 * DD_ + d], czi = o5[4 * DD_ + D_ + d];
    const float c_r = -czi, c_i = czr;
    const float up_r = arb[0] * g_r + arb[1] * mR[d] + arb[2] * v_r + arb[3] * s_r + arb[4] * c_r;
    const float up_i = arb[0] * g_i + arb[1] * mI[d] + arb[2] * v_i + arb[3] * s_i + arb[4] * c_i;
    gbuf[(size_t)b * DD_ + d]      = 0.6f * crd + 0.4f * up_r;
    gbuf[(size_t)b * DD_ + D_ + d] = 0.6f * cid + 0.4f * up_i;
  }
}

// ---------------------------------------------------------------------------
// Memory write + per-slot LayerNorm (in place on the ws copies)
// ---------------------------------------------------------------------------
__global__ __launch_bounds__(128) void mem_kernel(
    float* __restrict__ memR, float* __restrict__ memI,
    const float* __restrict__ zf, const float* __restrict__ eff,
    const float* __restrict__ gr_g, const float* __restrict__ gr_b,
    const float* __restrict__ gi_g, const float* __restrict__ gi_b)
{
  __shared__ float red[8];
  const int b = blockIdx.x >> 5, s = blockIdx.x & 31, t = threadIdx.x;
  const float e = eff[b * SLOTS_ + s];
  const float om = 1.f - e;
  const size_t base = ((size_t)b * SLOTS_ + s) * D_;
#pragma unroll
  for (int half = 0; half < 2; ++half) {
    float* mem = half ? memI : memR;
    const float* c  = zf + (size_t)b * DD_ + half * D_;
    const float* gg = half ? gi_g : gr_g;
    const float* bb = half ? gi_b : gr_b;
    float v[4];
#pragma unroll
    for (int j = 0; j < 4; ++j) {
      const int d = t + j * 128;
      v[j] = om * mem[base + d] + e * c[d];
    }
    const float ss = blockReduceSum(v[0] + v[1] + v[2] + v[3], red);
    const float mu = ss * (1.f / D_);
    float s2l = 0.f;
#pragma unroll
    for (int j = 0; j < 4; ++j) { v[j] -= mu; s2l += v[j] * v[j]; }
    const float s2 = blockReduceSum(s2l, red);
    const float inv = rsqrtf(s2 * (1.f / D_) + 1e-5f);
#pragma unroll
    for (int j = 0; j < 4; ++j) {
      const int d = t + j * 128;
      mem[base + d] = v[j] * inv * gg[d] + bb[d];
    }
  }
}

// ---------------------------------------------------------------------------
extern "C" void kernel_launch(void* const* d_in, const int* in_sizes, int n_in,
                              void* d_out, int out_size, void* d_ws, size_t ws_size,
                              hipStream_t stream)
{
  (void)in_sizes; (void)n_in; (void)out_size; (void)ws_size;
  const float* xR  = (const float*)d_in[0];
  const float* xI  = (const float*)d_in[1];
  const float* gwR = (const float*)d_in[2];
  const float* gwI = (const float*)d_in[3];
  const float* memR_in = (const float*)d_in[4];
  const float* memI_in = (const float*)d_in[5];
  const float* norm_r_g = (const float*)d_in[6];
  const float* norm_r_b = (const float*)d_in[7];
  const float* norm_i_g = (const float*)d_in[8];
  const float* norm_i_b = (const float*)d_in[9];
  const float* mem_norm_r_g = (const float*)d_in[10];
  const float* mem_norm_r_b = (const float*)d_in[11];
  const float* mem_norm_i_g = (const float*)d_in[12];
  const float* mem_norm_i_b = (const float*)d_in[13];
  // d_in[14..33]: {q,k,v,proj,critic} x {Wr,br,Wi,bi}
  const float* gateW = (const float*)d_in[34];
  const float* gateB = (const float*)d_in[35];
  const float* addrW = (const float*)d_in[36];
  const float* addrB = (const float*)d_in[37];
  const float* vqE   = (const float*)d_in[38];
  const float* visP  = (const float*)d_in[39];
  const float* audP  = (const float*)d_in[40];
  const float* arbW  = (const float*)d_in[41];
  const float* arbB  = (const float*)d_in[42];
  const float* decW  = (const float*)d_in[43];
  const float* decB  = (const float*)d_in[44];
  const float* inputGate = (const float*)d_in[45];

  float* ws = (float*)d_ws;
  size_t o = 0;
  float* gbuf  = ws + o; o += (size_t)B_ * DD_;           // [gr|gi] per row
  float* zfb   = ws + o; o += (size_t)B_ * DD_;           // [cr|ci] per row
  float* out5  = ws + o; o += (size_t)B_ * NCLIN_;        // q,k,v,proj,critic
  float* vqb   = ws + o; o += (size_t)B_ * DD_;
  float* Wclin = ws + o; o += (size_t)NCLIN_ * DD_;
  float* bclin = ws + o; o += NCLIN_;
  float* enorm = ws + o; o += KC_;
  float* effb  = ws + o; o += (size_t)B_ * SLOTS_;
  float* memR  = ws + o; o += (size_t)B_ * SLOTS_ * D_;
  float* memI  = ws + o; o += (size_t)B_ * SLOTS_ * D_;

  const size_t memBytes = (size_t)B_ * SLOTS_ * D_ * sizeof(float);
  hipMemcpyAsync(memR, memR_in, memBytes, hipMemcpyDeviceToDevice, stream);
  hipMemcpyAsync(memI, memI_in, memBytes, hipMemcpyDeviceToDevice, stream);

  prep_input<<<(B_ * D_ + 255) / 256, 256, 0, stream>>>(xR, xI, gwR, gwI, inputGate, gbuf);
  for (int c = 0; c < 5; ++c) {
    const float* Wr = (const float*)d_in[14 + 4 * c + 0];
    const float* br = (const float*)d_in[14 + 4 * c + 1];
    const float* Wi = (const float*)d_in[14 + 4 * c + 2];
    const float* bi = (const float*)d_in[14 + 4 * c + 3];
    prep_wclin<<<DD_, 256, 0, stream>>>(Wr, br, Wi, bi,
                                        Wclin + (size_t)c * DD_ * DD_, bclin + c * DD_);
  }
  prep_enorm<<<1, 128, 0, stream>>>(vqE, enorm);

  for (int it = 0; it < RITER_; ++it) {
    ln2_kernel<<<B_, 256, 0, stream>>>(gbuf, norm_r_g, norm_r_b, norm_i_g, norm_i_b, zfb);
    gemm_f32_wmma<<<dim3(B_ / 64, NCLIN_ / 128), 256, 0, stream>>>(
        zfb, Wclin, bclin, out5, B_, NCLIN_, DD_);
    vq_kernel<<<B_, 128, 0, stream>>>(zfb, vqE, enorm, vqb);
    row_kernel<<<B_, 256, 0, stream>>>(zfb, out5, vqb, memR, memI,
                                       gateW, gateB, addrW, addrB,
                                       visP, audP, arbW, arbB, gbuf, effb);
    mem_kernel<<<B_ * SLOTS_, 128, 0, stream>>>(memR, memI, zfb, effb,
                                                mem_norm_r_g, mem_norm_r_b,
                                                mem_norm_i_g, mem_norm_i_b);
  }

  // Decoder: logits = [gr|gi] @ dec_W.T + dec_b   (dominant GEMM)
  gemm_f32_wmma<<<dim3(B_ / 64, (VV_ + 127) / 128), 256, 0, stream>>>(
      gbuf, decW, decB, (float*)d_out, B_, VV_, DD_);
}